// HybridVQC_75780402971316
// MI455X (gfx1250) — compile-verified
//
#include <hip/hip_runtime.h>

typedef float v2f __attribute__((ext_vector_type(2)));
typedef float v8f __attribute__((ext_vector_type(8)));

#define NQ 7
#define NLAYERS 5
#define WAVES 4
#define PSI_STRIDE 132   // 16 rows x 132 floats, conflict-free b64 A-frag loads

// ---------------------------------------------------------------------------
// Kernel A: build the 128x128 entangling-layer unitary U in ws and repack it
// into a WMMA B-fragment-friendly layout.
// ws float layout:
//   [0     ,16384) packed Re(U): ((T*64+p)*16+n)*2+e = U[T*16+n][2p+e]
//   [16384 ,32768) packed Im(U)
//   [32768 ,49152) scratch Mre row-major  [i*128+j]
//   [49152 ,65536) scratch Mim
// ---------------------------------------------------------------------------
__global__ __launch_bounds__(256) void vqc_build_unitary(
    const float* __restrict__ qw, float* __restrict__ ws) {
  const int tid = threadIdx.x;
  float* Mre = ws + 32768;
  float* Mim = ws + 49152;
  for (int t = tid; t < 16384; t += 256) {
    Mre[t] = ((t >> 7) == (t & 127)) ? 1.0f : 0.0f;
    Mim[t] = 0.0f;
  }
  __threadfence();
  __syncthreads();

  for (int l = 0; l < NLAYERS; ++l) {
    const int r = l % (NQ - 1) + 1;
    // Rot(phi, theta, omega) on each wire
    for (int q = 0; q < NQ; ++q) {
      const float phi = qw[(l*NQ + q)*3 + 0];
      const float th  = qw[(l*NQ + q)*3 + 1];
      const float om  = qw[(l*NQ + q)*3 + 2];
      const float c  = cosf(0.5f*th), s = sinf(0.5f*th);
      const float ap = 0.5f*(phi + om), am = 0.5f*(phi - om);
      const float epr = cosf(ap), epi = -sinf(ap);
      const float emr = cosf(am), emi =  sinf(am);
      const float u00r =  epr*c, u00i =  epi*c;
      const float u01r = -emr*s, u01i = -emi*s;
      const float u10r =  emr*s, u10i = -emi*s;
      const float u11r =  epr*c, u11i = -epi*c;
      const int b = 6 - q, mask = 1 << b;
      for (int t = tid; t < 8192; t += 256) {
        const int p  = t >> 7, cc = t & 127;
        const int i0 = ((p >> b) << (b + 1)) | (p & (mask - 1));
        const int a0 = (i0)        * 128 + cc;
        const int a1 = (i0 | mask) * 128 + cc;
        const float x0r = Mre[a0], x0i = Mim[a0];
        const float x1r = Mre[a1], x1i = Mim[a1];
        Mre[a0] = u00r*x0r - u00i*x0i + u01r*x1r - u01i*x1i;
        Mim[a0] = u00r*x0i + u00i*x0r + u01r*x1i + u01i*x1r;
        Mre[a1] = u10r*x0r - u10i*x0i + u11r*x1r - u11i*x1i;
        Mim[a1] = u10r*x0i + u10i*x0r + u11r*x1i + u11i*x1r;
      }
      __threadfence();
      __syncthreads();
    }
    // ring of CNOT(q, q+r): pure row permutation
    for (int q = 0; q < NQ; ++q) {
      const int bc = 6 - q, bt = 6 - ((q + r) % NQ);
      const int lo = bc < bt ? bc : bt;
      const int hb = bc < bt ? bt : bc;
      for (int t = tid; t < 4096; t += 256) {
        const int p = t >> 7, cc = t & 127;
        int tmp = ((p   >> lo) << (lo + 1)) | (p   & ((1 << lo) - 1));
        tmp     = ((tmp >> hb) << (hb + 1)) | (tmp & ((1 << hb) - 1));
        const int i0 = tmp | (1 << bc);
        const int a0 = i0 * 128 + cc;
        const int a1 = (i0 | (1 << bt)) * 128 + cc;
        float tr = Mre[a0]; Mre[a0] = Mre[a1]; Mre[a1] = tr;
        float ti = Mim[a0]; Mim[a0] = Mim[a1]; Mim[a1] = ti;
      }
      __threadfence();
      __syncthreads();
    }
  }
  // repack: B[k][n] = U[n_global][k], (k,k+1) pairs contiguous per n
  for (int t = tid; t < 16384; t += 256) {
    const int e = t & 1;
    const int n = (t >> 1) & 15;
    const int p = (t >> 5) & 63;
    const int T = t >> 11;
    const int src = (T*16 + n)*128 + (2*p + e);
    ws[t]         = Mre[src];
    ws[16384 + t] = Mim[src];
  }
}

// ---------------------------------------------------------------------------
// Kernel B: per wave, 16 batch rows. psi0 in LDS -> f32 WMMA GEMM vs U
// (Re & Im chains) -> |psi|^2 with PauliZ signs -> shfl reduce -> MLP.
// ---------------------------------------------------------------------------
__global__ __launch_bounds__(WAVES*32) void vqc_forward(
    const float* __restrict__ x,  const float* __restrict__ uPk,
    const float* __restrict__ W1, const float* __restrict__ b1,
    const float* __restrict__ W2, const float* __restrict__ b2,
    float* __restrict__ out) {
  __shared__ float smem[WAVES*16*PSI_STRIDE + WAVES*224];
  const int tid  = threadIdx.x;
  const int wave = tid >> 5;
  const int lane = tid & 31;
  const int m    = lane & 15;
  const int hi   = lane >> 4;
  const int b0   = blockIdx.x * (WAVES*16) + wave*16;
  float* psi = smem + wave*16*PSI_STRIDE;
  float* cs  = smem + WAVES*16*PSI_STRIDE + wave*224;

  // 1) cos/sin of x/2 for this wave's 16 rows
  for (int t = lane; t < 112; t += 32) {
    const float a = 0.5f * x[(b0 + t/7)*7 + (t % 7)];
    cs[t*2 + 0] = cosf(a);
    cs[t*2 + 1] = sinf(a);
  }
  __syncthreads();

  // 2) product state psi0[row][j] (real)
  for (int row = 0; row < 16; ++row) {
    float cv[7], sv[7];
    #pragma unroll
    for (int q = 0; q < 7; ++q) {
      cv[q] = cs[(row*7 + q)*2 + 0];
      sv[q] = cs[(row*7 + q)*2 + 1];
    }
    #pragma unroll
    for (int k = 0; k < 4; ++k) {
      const int j = k*32 + lane;
      float v = 1.0f;
      #pragma unroll
      for (int q = 0; q < 7; ++q) v *= ((j >> (6 - q)) & 1) ? sv[q] : cv[q];
      psi[row*PSI_STRIDE + j] = v;
    }
  }
  __syncthreads();

  // 3) psi = U psi0 via V_WMMA_F32_16X16X4_F32; fold PauliZ signs on the fly
  float slo[7];
  slo[0] = slo[1] = slo[2] = 1.0f;
  #pragma unroll
  for (int q = 3; q < 7; ++q) slo[q] = 1.0f - 2.0f*(float)((m >> (6 - q)) & 1);
  float zacc[56];
  #pragma unroll
  for (int i = 0; i < 56; ++i) zacc[i] = 0.0f;

  for (int T = 0; T < 8; ++T) {
    v8f accR = {}; v8f accI = {};
    const float* bbase = uPk + T*2048;        // T*64 pair-slots * 16 n * 2
    #pragma unroll 4
    for (int kk = 0; kk < 32; ++kk) {         // K = 128 in steps of 4
      const v2f a  = *(const v2f*)(psi + m*PSI_STRIDE + kk*4 + 2*hi);
      const float* bp = bbase + ((2*kk + hi)*16 + m)*2;
      const v2f br = *(const v2f*)(bp);
      const v2f bi = *(const v2f*)(bp + 16384);
      accR = __builtin_amdgcn_wmma_f32_16x16x4_f32(
          false, a, false, br, (short)0, accR, false, false);
      accI = __builtin_amdgcn_wmma_f32_16x16x4_f32(
          false, a, false, bi, (short)0, accI, false, false);
    }
    float sg[7];
    sg[0] = (T & 4) ? -1.0f : 1.0f;           // wire0 <-> bit6 of n (tile bit 2)
    sg[1] = (T & 2) ? -1.0f : 1.0f;
    sg[2] = (T & 1) ? -1.0f : 1.0f;
    sg[3] = slo[3]; sg[4] = slo[4]; sg[5] = slo[5]; sg[6] = slo[6];
    #pragma unroll
    for (int v = 0; v < 8; ++v) {
      const float pv = accR[v]*accR[v] + accI[v]*accI[v];
      #pragma unroll
      for (int q = 0; q < 7; ++q) zacc[q*8 + v] += sg[q]*pv;
    }
  }

  // 4) reduce across the 16 lanes of each half-wave (rows m and m+8)
  #pragma unroll
  for (int st = 0; st < 4; ++st) {
    const int d = 1 << st;
    #pragma unroll
    for (int i = 0; i < 56; ++i) zacc[i] += __shfl_xor(zacc[i], d, 32);
  }

  // 5) stash q_out[16][7] in (now free) psi region
  if (m < 7) {
    #pragma unroll
    for (int v = 0; v < 8; ++v) psi[(v + 8*hi)*7 + m] = zacc[m*8 + v];
  }
  __syncthreads();

  // 6) h = relu(q_out @ W1^T + b1)  -> psi[128 + row*16 + u]
  for (int t = lane; t < 256; t += 32) {
    const int row = t >> 4, u = t & 15;
    float acc = b1[u];
    #pragma unroll
    for (int q = 0; q < 7; ++q) acc += psi[row*7 + q] * W1[u*7 + q];
    psi[128 + t] = fmaxf(acc, 0.0f);
  }
  __syncthreads();

  // 7) logits = h @ W2^T + b2
  for (int t = lane; t < 352; t += 32) {
    const int row = t / 22, cls = t % 22;
    float acc = b2[cls];
    #pragma unroll
    for (int u = 0; u < 16; ++u) acc += psi[128 + row*16 + u] * W2[cls*16 + u];
    out[(b0 + row)*22 + cls] = acc;
  }
}

extern "C" void kernel_launch(void* const* d_in, const int* in_sizes, int n_in,
                              void* d_out, int out_size, void* d_ws, size_t ws_size,
                              hipStream_t stream) {
  const float* x  = (const float*)d_in[0];
  const float* qw = (const float*)d_in[1];
  const float* W1 = (const float*)d_in[2];
  const float* b1 = (const float*)d_in[3];
  const float* W2 = (const float*)d_in[4];
  const float* b2 = (const float*)d_in[5];
  float* out = (float*)d_out;
  float* ws  = (float*)d_ws;           // needs 65536 floats = 256 KB

  const int B = in_sizes[0] / NQ;      // 32768

  vqc_build_unitary<<<1, 256, 0, stream>>>(qw, ws);
  vqc_forward<<<B / (WAVES*16), WAVES*32, 0, stream>>>(x, ws, W1, b1, W2, b2, out);
}